// AGCN_62457414419111
// MI455X (gfx1250) — compile-verified
//
#include <hip/hip_runtime.h>
#include <hip/hip_bf16.h>
#include <math.h>

typedef __attribute__((ext_vector_type(2))) float v2f;
typedef __attribute__((ext_vector_type(8))) float v8f;
typedef __attribute__((ext_vector_type(4))) unsigned int u32x4;
typedef __attribute__((ext_vector_type(8))) int i32x8;
typedef __attribute__((ext_vector_type(4))) int i32x4;

#define DFEAT 48
#define KSTACK 144            // 3 * DFEAT stacked Chebyshev K-dim
#define WPAD 146              // padded K stride (bank-conflict-free, 8B aligned)
#define BN_EPS 1e-5f

#if defined(__has_builtin)
#if __has_builtin(__builtin_amdgcn_tensor_load_to_lds) && \
    __has_builtin(__builtin_amdgcn_s_wait_tensorcnt)
#define TDM_OK 1
#endif
#endif

// ---------------------------------------------------------------------------
// float atomic max via monotonic int-bits trick (handles negatives, -INF init)
__device__ __forceinline__ void atomicMaxF(float* addr, float val) {
  if (val >= 0.0f) atomicMax((int*)addr, __float_as_int(val));
  else             atomicMin((unsigned int*)addr, __float_as_uint(val));
}

// ---------------------------------------------------------------------------
__global__ void fill_kernel(float* __restrict__ p, float v, long long n) {
  long long i = (long long)blockIdx.x * blockDim.x + threadIdx.x;
  if (i < n) p[i] = v;
}

__global__ void degree_kernel(const int* __restrict__ row, float* __restrict__ deg, int E) {
  int e = blockIdx.x * blockDim.x + threadIdx.x;
  if (e < E) atomicAdd(&deg[row[e]], 1.0f);
}

__global__ void dinv_kernel(const float* __restrict__ deg, float* __restrict__ dinv, int N) {
  int i = blockIdx.x * blockDim.x + threadIdx.x;
  if (i < N) {
    float d = deg[i];
    dinv[i] = (d > 0.0f) ? rsqrtf(fmaxf(d, 1.0f)) : 0.0f;
  }
}

__global__ void norm_kernel(const int* __restrict__ row, const int* __restrict__ col,
                            const float* __restrict__ dinv, float* __restrict__ nrm, int E) {
  int e = blockIdx.x * blockDim.x + threadIdx.x;
  if (e < E) nrm[e] = -dinv[row[e]] * dinv[col[e]];
}

// L_hat sparse matvec: out[row] += norm * in[col]  (coalesced over feature dim)
__global__ void lhat_kernel(const int* __restrict__ row, const int* __restrict__ col,
                            const float* __restrict__ nrm,
                            const float* __restrict__ in, float* __restrict__ out,
                            long long total) {
  long long i = (long long)blockIdx.x * blockDim.x + threadIdx.x;
  if (i >= total) return;
  int e = (int)(i / DFEAT);
  int d = (int)(i - (long long)e * DFEAT);
  atomicAdd(&out[(size_t)row[e] * DFEAT + d], nrm[e] * in[(size_t)col[e] * DFEAT + d]);
}

// ---------------------------------------------------------------------------
// Precompute stacked+folded ChebConv weights, transposed & padded:
//   wsT[o * WPAD + k] = { W0-W2 | W1 | 2*W2 }[k][o],  k in [0,144), pad -> 0
// so  t = cur@(W0-W2) + tx1@W1 + L(tx1)@(2*W2) + b  ==  ChebConv(K=3).
__global__ void wstack_kernel(const float* __restrict__ Wk, float* __restrict__ wsT) {
  int i = blockIdx.x * blockDim.x + threadIdx.x;
  if (i >= DFEAT * WPAD) return;
  int o = i / WPAD, k = i - o * WPAD;
  float v = 0.0f;
  if (k < KSTACK) {
    int s = k / DFEAT, r = k - s * DFEAT;
    if (s == 0)      v = Wk[r * DFEAT + o] - Wk[2 * DFEAT * DFEAT + r * DFEAT + o];
    else if (s == 1) v = Wk[DFEAT * DFEAT + r * DFEAT + o];
    else             v = 2.0f * Wk[2 * DFEAT * DFEAT + r * DFEAT + o];
  }
  wsT[i] = v;
}

// ---------------------------------------------------------------------------
// Fused ChebConv GEMM (WMMA f32 16x16x4) + bias + ReLU + BN sum/sumsq stats.
// Weights staged LDS via Tensor Data Mover (one D# 2-D tile per workgroup).
__global__ __launch_bounds__(256) void cheb_gemm_wmma(
    const float* __restrict__ cur, const float* __restrict__ tx1,
    const float* __restrict__ tx2, const float* __restrict__ wsT,
    const float* __restrict__ bias, float* __restrict__ tout,
    float* __restrict__ stats, int N) {
  __shared__ float wlds[DFEAT * WPAD];

#if TDM_OK
  if (threadIdx.x < 32) {   // one wave issues the TDM transfer (EXEC ignored)
    unsigned long long ga = (unsigned long long)(uintptr_t)wsT;
    unsigned ldsOff = (unsigned)(uintptr_t)(void*)wlds;   // addr[31:0] = LDS offset
    // D# group0: count=1, lds_addr, global_addr[56:0], type=2 (ISA 8.3)
    u32x4 g0;
    g0[0] = 1u;
    g0[1] = ldsOff;
    g0[2] = (unsigned)(ga & 0xFFFFFFFFu);
    g0[3] = (unsigned)((ga >> 32) & 0x1FFFFFFu) | (2u << 30);
    // D# group1: data_size=4B; tensor 146x48; tile 146x48; stride0=146 (ISA 8.4)
    i32x8 g1;
    g1[0] = 0x20000;                 // data_size=2 (4 bytes), mask=0
    g1[1] = (int)(WPAD << 16);       // tensor_dim0[15:0] in bits 63:48
    g1[2] = (int)(DFEAT << 16);      // tensor_dim0 hi=0 | tensor_dim1[15:0]
    g1[3] = (int)(WPAD << 16);       // tensor_dim1 hi=0 | tile_dim0
    g1[4] = DFEAT;                   // tile_dim1 | tile_dim2=0
    g1[5] = WPAD;                    // tensor_dim0_stride[31:0]
    g1[6] = 0;
    g1[7] = 0;
    i32x4 g2 = {0, 0, 0, 0};
    i32x4 g3 = {0, 0, 0, 0};
#if __clang_major__ >= 23
    i32x8 g4 = {0, 0, 0, 0, 0, 0, 0, 0};
    __builtin_amdgcn_tensor_load_to_lds(g0, g1, g2, g3, g4, 0);
#else
    __builtin_amdgcn_tensor_load_to_lds(g0, g1, g2, g3, 0);
#endif
    __builtin_amdgcn_s_wait_tensorcnt(0);
  }
#else
  for (int i = threadIdx.x; i < DFEAT * WPAD; i += 256) wlds[i] = wsT[i];
#endif
  __syncthreads();

  const int lane = threadIdx.x & 31;
  const int wave = threadIdx.x >> 5;
  const int numTiles = (N + 15) >> 4;
  const int tile = blockIdx.x * 8 + wave;
  if (tile >= numTiles) return;          // uniform per wave: EXEC stays all-1s

  const int m = lane & 15;               // row-in-tile (A) / col (B,C)
  const int h = lane >> 4;               // half select
  const int row0 = tile << 4;
  int arow = row0 + m;
  if (arow >= N) arow = N - 1;           // clamp (stores are guarded below)

  v8f acc0 = {0, 0, 0, 0, 0, 0, 0, 0};
  v8f acc1 = {0, 0, 0, 0, 0, 0, 0, 0};
  v8f acc2 = {0, 0, 0, 0, 0, 0, 0, 0};

  const float* wbase = wlds + m * WPAD + 2 * h;   // B frag: b[j]=B[k0+2h+j][o]

  for (int kt = 0; kt < 36; ++kt) {      // K = 144, 4 per step
    const int k0 = kt * 4;
    const float* src = (kt < 12) ? cur : (kt < 24 ? tx1 : tx2);
    const int kloc = (k0 < 48) ? k0 : (k0 < 96 ? k0 - 48 : k0 - 96);

    // A frag: a[j] = A[m][2h + j]  (ISA 32-bit A 16x4 layout), 8B load
    v2f a = *(const v2f*)(src + (size_t)arow * DFEAT + kloc + 2 * h);

    // B frags: contiguous 8B ds_load_b64 each (transposed, padded layout)
    const float* wp = wbase + k0;
    v2f b0  = *(const v2f*)(wp);
    v2f b1v = *(const v2f*)(wp + 16 * WPAD);
    v2f b2v = *(const v2f*)(wp + 32 * WPAD);

    acc0 = __builtin_amdgcn_wmma_f32_16x16x4_f32(false, a, false, b0,  (short)0, acc0, false, false);
    acc1 = __builtin_amdgcn_wmma_f32_16x16x4_f32(false, a, false, b1v, (short)0, acc1, false, false);
    acc2 = __builtin_amdgcn_wmma_f32_16x16x4_f32(false, a, false, b2v, (short)0, acc2, false, false);
  }

  // Epilogue: bias + ReLU + store + BN partial sums (per output feature o).
  v8f accs[3] = {acc0, acc1, acc2};
#pragma unroll
  for (int nt = 0; nt < 3; ++nt) {
    const int o = nt * 16 + m;
    const float bi = bias[o];
    float s = 0.0f, q = 0.0f;
#pragma unroll
    for (int r = 0; r < 8; ++r) {        // C/D layout: c[r] = C[r + 8h][m]
      const int rr = row0 + r + 8 * h;
      float v = accs[nt][r] + bi;
      v = v > 0.0f ? v : 0.0f;           // ReLU
      if (rr < N) {
        tout[(size_t)rr * DFEAT + o] = v;
        s += v;
        q += v * v;
      }
    }
    s += __shfl_xor(s, 16);              // fold the two halves (same o)
    q += __shfl_xor(q, 16);
    if (h == 0) {
      atomicAdd(&stats[o], s);
      atomicAdd(&stats[48 + o], q);
    }
  }
}

// ---------------------------------------------------------------------------
__global__ void bn_finalize_kernel(float* __restrict__ stats,
                                   const float* __restrict__ gamma,
                                   const float* __restrict__ beta, int N) {
  int o = threadIdx.x;
  if (o < DFEAT) {
    float invN = 1.0f / (float)N;
    float mean = stats[o] * invN;
    float var  = stats[48 + o] * invN - mean * mean;
    float sc   = gamma[o] * rsqrtf(var + BN_EPS);
    stats[96 + o]  = sc;                  // scale
    stats[144 + o] = beta[o] - mean * sc; // shift
  }
}

// graph_max_pool with fused BN apply: out[row] = max(out[row], sc*t[col]+sh)
__global__ void pool_kernel(const int* __restrict__ row, const int* __restrict__ col,
                            const float* __restrict__ t, const float* __restrict__ stats,
                            float* __restrict__ outp, long long total) {
  long long i = (long long)blockIdx.x * blockDim.x + threadIdx.x;
  if (i >= total) return;
  int e = (int)(i / DFEAT);
  int d = (int)(i - (long long)e * DFEAT);
  float v = stats[96 + d] * t[(size_t)col[e] * DFEAT + d] + stats[144 + d];
  atomicMaxF(&outp[(size_t)row[e] * DFEAT + d], v);
}

__global__ void neginf_fix_kernel(float* __restrict__ p, long long n) {
  long long i = (long long)blockIdx.x * blockDim.x + threadIdx.x;
  if (i < n) {
    float v = p[i];
    p[i] = (isinf(v) && v < 0.0f) ? 0.0f : v;
  }
}

__global__ void add_pool_kernel(const int* __restrict__ batch, const float* __restrict__ cur,
                                float* __restrict__ g, long long total) {
  long long i = (long long)blockIdx.x * blockDim.x + threadIdx.x;
  if (i >= total) return;
  int n = (int)(i / DFEAT);
  int d = (int)(i - (long long)n * DFEAT);
  atomicAdd(&g[(size_t)batch[n] * DFEAT + d], cur[i]);
}

__global__ void mlp1_kernel(const float* __restrict__ g, const float* __restrict__ W1,
                            const float* __restrict__ b1, float* __restrict__ hbuf,
                            int G, int H) {
  int idx = blockIdx.x * blockDim.x + threadIdx.x;
  if (idx >= G * H) return;
  int gi = idx / H, t = idx - gi * H;
  float acc = b1[t];
  for (int d = 0; d < DFEAT; ++d) acc += g[gi * DFEAT + d] * W1[d * H + t];
  hbuf[idx] = acc > 0.0f ? acc : 0.0f;
}

__global__ void mlp2_kernel(const float* __restrict__ hbuf, const float* __restrict__ W2,
                            const float* __restrict__ b2, float* __restrict__ out,
                            int G, int H, int O) {
  int idx = blockIdx.x * blockDim.x + threadIdx.x;
  if (idx >= G * O) return;
  int gi = idx / O, o = idx - gi * O;
  float acc = b2[o];
  for (int t = 0; t < H; ++t) acc += hbuf[gi * H + t] * W2[t * O + o];
  out[idx] = acc;
}

// ---------------------------------------------------------------------------
extern "C" void kernel_launch(void* const* d_in, const int* in_sizes, int n_in,
                              void* d_out, int out_size, void* d_ws, size_t ws_size,
                              hipStream_t stream) {
  const float* x     = (const float*)d_in[0];
  const int*   ei    = (const int*)d_in[1];
  const int*   batch = (const int*)d_in[2];
  const float* Wk    = (const float*)d_in[4];
  const float* b     = (const float*)d_in[5];
  const float* gamma = (const float*)d_in[6];
  const float* beta  = (const float*)d_in[7];
  const float* W1    = (const float*)d_in[8];
  const float* b1    = (const float*)d_in[9];
  const float* W2    = (const float*)d_in[10];
  const float* b2    = (const float*)d_in[11];

  const int N = in_sizes[0] / DFEAT;
  const int E = in_sizes[1] / 2;
  const int G = 64, H = 128, O = 12;
  const int* row = ei;
  const int* col = ei + E;

  // workspace carve-up (floats), 256B aligned chunks
  float* ws = (float*)d_ws;
  size_t off = 0;
  auto carve = [&](size_t nf) { float* p = ws + off; off += (nf + 63) & ~(size_t)63; return p; };
  float* deg   = carve((size_t)N);
  float* dinv  = carve((size_t)N);
  float* nrm   = carve((size_t)E);
  float* cur   = carve((size_t)N * DFEAT);
  float* tx1   = carve((size_t)N * DFEAT);
  float* tx2   = carve((size_t)N * DFEAT);
  float* tbuf  = carve((size_t)N * DFEAT);
  float* wsT   = carve((size_t)DFEAT * WPAD);
  float* stats = carve(192);
  float* gbuf  = carve((size_t)G * DFEAT);
  float* hbuf  = carve((size_t)G * H);

  const long long nd = (long long)N * DFEAT;
  const long long ed = (long long)E * DFEAT;
  const int TB = 256;
  auto blk = [&](long long n) { return (unsigned)((n + TB - 1) / TB); };

  // edge normalization + folded weights (constant across iterations)
  fill_kernel<<<blk(N), TB, 0, stream>>>(deg, 0.0f, N);
  degree_kernel<<<blk(E), TB, 0, stream>>>(row, deg, E);
  dinv_kernel<<<blk(N), TB, 0, stream>>>(deg, dinv, N);
  norm_kernel<<<blk(E), TB, 0, stream>>>(row, col, dinv, nrm, E);
  wstack_kernel<<<blk(DFEAT * WPAD), TB, 0, stream>>>(Wk, wsT);

  hipMemcpyAsync(cur, x, (size_t)nd * sizeof(float), hipMemcpyDeviceToDevice, stream);

  const int numTiles = (N + 15) / 16;
  const unsigned gemmBlocks = (unsigned)((numTiles + 7) / 8);

  for (int it = 0; it < 5; ++it) {
    // Tx1 = L(cur)
    fill_kernel<<<blk(nd), TB, 0, stream>>>(tx1, 0.0f, nd);
    lhat_kernel<<<blk(ed), TB, 0, stream>>>(row, col, nrm, cur, tx1, ed);
    // tx2 = L(tx1)   (the 2x and -Tx0 are folded into the stacked weights)
    fill_kernel<<<blk(nd), TB, 0, stream>>>(tx2, 0.0f, nd);
    lhat_kernel<<<blk(ed), TB, 0, stream>>>(row, col, nrm, tx1, tx2, ed);
    // fused ChebConv GEMM + ReLU + BN stats
    fill_kernel<<<1, TB, 0, stream>>>(stats, 0.0f, 192);
    cheb_gemm_wmma<<<gemmBlocks, TB, 0, stream>>>(cur, tx1, tx2, wsT, b, tbuf, stats, N);
    bn_finalize_kernel<<<1, 64, 0, stream>>>(stats, gamma, beta, N);
    // graph max pool (BN applied on the fly), empty segments -> 0
    fill_kernel<<<blk(nd), TB, 0, stream>>>(cur, -INFINITY, nd);
    pool_kernel<<<blk(ed), TB, 0, stream>>>(row, col, tbuf, stats, cur, ed);
    neginf_fix_kernel<<<blk(nd), TB, 0, stream>>>(cur, nd);
  }

  // global add pool + MLP head
  fill_kernel<<<blk(G * DFEAT), TB, 0, stream>>>(gbuf, 0.0f, G * DFEAT);
  add_pool_kernel<<<blk(nd), TB, 0, stream>>>(batch, cur, gbuf, nd);
  mlp1_kernel<<<blk(G * H), TB, 0, stream>>>(gbuf, W1, b1, hbuf, G, H);
  mlp2_kernel<<<blk(G * O), TB, 0, stream>>>(hbuf, W2, b2, (float*)d_out, G, H, O);
}